// BilateralVolumetricRenderer_36816459662064
// MI455X (gfx1250) — compile-verified
//
#include <hip/hip_runtime.h>
#include <stdint.h>

#define SAMPLES 128
#define WAVES_PER_BLOCK 4
#define THREADS (WAVES_PER_BLOCK * 32)

__device__ __forceinline__ float bperm_f(int srcLane, float v) {
    // dst[lane] = v[srcLane]; index in bytes, bits[6:2] used (wraps) on wave32
    return __int_as_float(__builtin_amdgcn_ds_bpermute(srcLane << 2, __float_as_int(v)));
}

__global__ __launch_bounds__(THREADS) void bilateral_volrend_kernel(
    const float* __restrict__ rgb,    // [N, 128, 3]
    const float* __restrict__ nbr,    // [5, N, 128, 3] (flattened 5N rows)
    const float* __restrict__ sigma,  // [N, 128]
    const float* __restrict__ zvals,  // [N, 128]
    const int*   __restrict__ wbkgd,  // scalar
    float* __restrict__ comp,         // [N, 3]
    float* __restrict__ depth,        // [N]
    float* __restrict__ opac,         // [N]
    float* __restrict__ wout,         // [N, 128]
    int n_ray)
{
    // per-wave LDS: 384 floats rgb + 5*384 floats neighbors = 9216 B
    __shared__ __align__(16) float sh[WAVES_PER_BLOCK][6 * 384];

    const int lane = threadIdx.x & 31;
    const int wid  = threadIdx.x >> 5;
    const int ray  = blockIdx.x * WAVES_PER_BLOCK + wid;
    if (ray >= n_ray) return;

    float* shw = &sh[wid][0];
    const unsigned lbase = (unsigned)(uintptr_t)shw;  // flat LDS addr low 32 bits == LDS offset

    // ---------------- issue async global -> LDS bulk copies (CDNA5 ASYNCcnt path) ------------
    {
        const float* g = rgb + (size_t)ray * (3 * SAMPLES);
        #pragma unroll
        for (int j = 0; j < 3; ++j) {
            const float* ga = g + (j * 32 + lane) * 4;               // 16B per lane
            unsigned la = lbase + (unsigned)((j * 32 + lane) * 16);
            asm volatile("global_load_async_to_lds_b128 %0, %1, off"
                         :: "v"(la), "v"(ga) : "memory");
        }
        #pragma unroll
        for (int k = 0; k < 5; ++k) {
            const float* gk = nbr + ((size_t)k * (size_t)n_ray + (size_t)ray) * (3 * SAMPLES);
            #pragma unroll
            for (int j = 0; j < 3; ++j) {
                const float* ga = gk + (j * 32 + lane) * 4;
                unsigned la = lbase + (unsigned)((384u * (k + 1) + (j * 32 + lane) * 4) * 4);
                asm volatile("global_load_async_to_lds_b128 %0, %1, off"
                             :: "v"(la), "v"(ga) : "memory");
            }
        }
    }

    // ---------------- Phase A: volume-rendering weights (cumprod scan), overlapped ----------
    float wreg[4], zreg[4];
    float carry = 1.0f;
    const float* zrow = zvals + (size_t)ray * SAMPLES;
    const float* srow = sigma + (size_t)ray * SAMPLES;
    #pragma unroll
    for (int c = 0; c < 4; ++c) {
        const int s = c * 32 + lane;
        float z0 = zrow[s];
        float z1 = zrow[(s < SAMPLES - 1) ? s + 1 : s];          // guarded (no OOB)
        float delta = (s < SAMPLES - 1) ? (z1 - z0) : 1e10f;
        float sg = srow[s];
        float e = __expf(-delta * fmaxf(sg, 0.0f));
        float alpha = 1.0f - e;
        float t = e + 1e-10f;                                    // 1 - alpha + eps

        // wave-level inclusive product scan of t
        float p = t;
        #pragma unroll
        for (int d = 1; d < 32; d <<= 1) {
            float up = bperm_f(lane - d, p);
            if (lane >= d) p *= up;
        }
        float excl = bperm_f(lane - 1, p);
        if (lane == 0) excl = 1.0f;
        float w = alpha * (carry * excl);
        carry *= __int_as_float(__builtin_amdgcn_readlane(__float_as_int(p), 31));

        wreg[c] = w;
        zreg[c] = z0;
        wout[(size_t)ray * SAMPLES + s] = w;                     // coalesced
    }

    // ---------------- drain async copies before touching LDS --------------------------------
    asm volatile("s_wait_asynccnt 0x0" ::: "memory");

    // ---------------- Phase B: bilateral blend + compositing ---------------------------------
    float c0 = 0.f, c1 = 0.f, c2 = 0.f, dep = 0.f, op = 0.f;
    #pragma unroll
    for (int c = 0; c < 4; ++c) {
        const int s3 = (c * 32 + lane) * 3;
        float r0 = shw[s3 + 0], r1 = shw[s3 + 1], r2 = shw[s3 + 2];
        float n0 = r0, n1 = r1, n2 = r2, den = 1.0f;
        #pragma unroll
        for (int k = 0; k < 5; ++k) {
            const int b = 384 * (k + 1) + s3;
            float a0 = shw[b + 0], a1 = shw[b + 1], a2 = shw[b + 2];
            float d0 = r0 - a0, d1 = r1 - a1, d2 = r2 - a2;
            float wk = __expf(-(d0 * d0 + d1 * d1 + d2 * d2));   // gamma = 1
            n0 += a0 * wk; n1 += a1 * wk; n2 += a2 * wk; den += wk;
        }
        float inv = __builtin_amdgcn_rcpf(den);                  // den >= 1
        float w = wreg[c];
        bool m = (w >= 0.01f);
        float p0 = m ? n0 * inv : r0;
        float p1 = m ? n1 * inv : r1;
        float p2 = m ? n2 * inv : r2;
        c0 += w * p0; c1 += w * p1; c2 += w * p2;
        dep += w * zreg[c];
        op  += w;
    }

    // ---------------- wave butterfly reductions ---------------------------------------------
    #pragma unroll
    for (int d = 16; d >= 1; d >>= 1) {
        c0  += bperm_f(lane ^ d, c0);
        c1  += bperm_f(lane ^ d, c1);
        c2  += bperm_f(lane ^ d, c2);
        dep += bperm_f(lane ^ d, dep);
        op  += bperm_f(lane ^ d, op);
    }
    if (lane == 0) {
        if (wbkgd[0]) { float b = 1.0f - op; c0 += b; c1 += b; c2 += b; }
        comp[(size_t)ray * 3 + 0] = c0;
        comp[(size_t)ray * 3 + 1] = c1;
        comp[(size_t)ray * 3 + 2] = c2;
        depth[ray] = dep;
        opac[ray]  = op;
    }
}

extern "C" void kernel_launch(void* const* d_in, const int* in_sizes, int n_in,
                              void* d_out, int out_size, void* d_ws, size_t ws_size,
                              hipStream_t stream) {
    const float* rgb   = (const float*)d_in[0];
    const float* nbr   = (const float*)d_in[1];
    const float* sigma = (const float*)d_in[2];
    const float* zvals = (const float*)d_in[3];
    const int*   wbk   = (const int*)d_in[4];

    const int n_ray = in_sizes[2] / SAMPLES;   // sigma is [N, 128]

    float* out   = (float*)d_out;
    float* comp  = out;                          // [N,3]
    float* depth = out + (size_t)3 * n_ray;      // [N]
    float* opac  = out + (size_t)4 * n_ray;      // [N]
    float* wout  = out + (size_t)5 * n_ray;      // [N,128]

    const int blocks = (n_ray + WAVES_PER_BLOCK - 1) / WAVES_PER_BLOCK;
    hipLaunchKernelGGL(bilateral_volrend_kernel, dim3(blocks), dim3(THREADS), 0, stream,
                       rgb, nbr, sigma, zvals, wbk, comp, depth, opac, wout, n_ray);
}